// Word2MatEncoder_4939212390967
// MI455X (gfx1250) — compile-verified
//
#include <hip/hip_runtime.h>

// ---------------------------------------------------------------------------
// Word2Mat encoder for MI455X (gfx1250, wave32).
// Per batch element: product of 64 gathered 28x28 f32 matrices, computed with
// v_wmma_f32_16x16x4_f32 on zero-padded 32x32 tiles held in LDS.
// Gather path: GLOBAL_LOAD_ASYNC_TO_LDS_B128 (ASYNCcnt), prefetched one full
// compute phase ahead.
// ---------------------------------------------------------------------------

typedef __attribute__((ext_vector_type(2))) float v2f;
typedef __attribute__((ext_vector_type(8))) float v8f;
typedef __attribute__((ext_vector_type(4))) float f4;

#define MAT    28
#define PAD    32
#define SEQ    64
#define EMB    784          // 28*28
#define NV4    196          // 784 / 4  (float4 elements per matrix)
#define ROWV4  7            // 28 / 4   (float4 elements per row)
#define WPB    4            // waves per block
#define BATCH  256

// Types matching the async-to-LDS builtin's parameters (per hipcc diagnostic:
// "__attribute__((__vector_size__(4 * sizeof(int)))) int __device__ *").
typedef int b128_t __attribute__((vector_size(16)));
typedef __attribute__((address_space(3))) b128_t lds_b128_t;
typedef __attribute__((address_space(3))) float  lds_float;

__device__ __forceinline__ v8f wmma_f32(v2f a, v2f b, v8f c) {
  // (neg_a, A, neg_b, B, c_mod, C, reuse_a, reuse_b)
  return __builtin_amdgcn_wmma_f32_16x16x4_f32(
      false, a, false, b, (short)0, c, false, false);
}

// One 16-byte async global->LDS transfer (per-lane addresses, ASYNCcnt).
__device__ __forceinline__ void async_copy16(const float* gsrc, float* ldst) {
#if __has_builtin(__builtin_amdgcn_global_load_async_to_lds_b128)
  __builtin_amdgcn_global_load_async_to_lds_b128(
      (b128_t*)gsrc, (lds_b128_t*)ldst, 0, 0);
#else
  unsigned lds_off = (unsigned)(__UINTPTR_TYPE__)(lds_float*)ldst;
  asm volatile("global_load_async_to_lds_b128 %0, %1, off"
               :: "v"(lds_off), "v"(gsrc) : "memory");
#endif
}

__device__ __forceinline__ void wait_async0() {
#if __has_builtin(__builtin_amdgcn_s_wait_asynccnt)
  __builtin_amdgcn_s_wait_asynccnt(0);
  asm volatile("" ::: "memory");   // compiler barrier: LDS now updated
#else
  asm volatile("s_wait_asynccnt 0" ::: "memory");
#endif
}

__device__ __forceinline__ void fence_ds0() {
  asm volatile("s_wait_dscnt 0" ::: "memory");
}

__global__ void __launch_bounds__(WPB * 32)
w2m_chain_kernel(const int* __restrict__ sent,
                 const float* __restrict__ lw,
                 float* __restrict__ out) {
  // Per-wave LDS: running product P + double-buffered incoming matrix.
  __shared__ float lds[WPB * 3 * PAD * PAD];

  const int lane = threadIdx.x & 31;
  const int wave = threadIdx.x >> 5;
  const int b    = blockIdx.x * WPB + wave;

  float* P  = &lds[wave * 3 * PAD * PAD];
  float* B0 = P  + PAD * PAD;
  float* B1 = B0 + PAD * PAD;

  // Zero all three 32x32 buffers once (pad region must stay zero; the
  // 28x28 compute never writes the pad cells again because padded rows/cols
  // of A and B are zero => padded rows/cols of C are zero).
  for (int i = lane; i < 3 * PAD * PAD / 4; i += 32) {
    *(f4*)&P[4 * i] = f4{0.f, 0.f, 0.f, 0.f};
  }
  // ds stores and async-LDS writes use different counters and are unordered:
  // make sure the zero-fill has landed before any async data arrives.
  fence_ds0();

  const int* srow = sent + b * SEQ;

  // Issue the async gather of one 28x28 matrix (row-major, 784 contiguous
  // floats) into a padded 32x32 LDS tile; 28 floats/row = 7 aligned b128.
  auto issue_mat = [&](float* dst, int s) {
    const float* src = lw + (size_t)srow[s] * EMB;
    for (int e4 = lane; e4 < NV4; e4 += 32) {
      int r  = e4 / ROWV4;
      int c4 = e4 % ROWV4;
      async_copy16(&src[4 * e4], &dst[r * PAD + 4 * c4]);
    }
  };

  issue_mat(P, 0);    // running product := M0
  issue_mat(B0, 1);   // first operand

  const int half = lane >> 4;   // lane group (0: lanes 0-15, 1: lanes 16-31)
  const int l16  = lane & 15;

  for (int s = 1; s < SEQ; ++s) {
    float* Bc = (s & 1) ? B0 : B1;   // current operand (issued last iter)
    float* Bn = (s & 1) ? B1 : B0;   // buffer for next operand

    // Operand for this step (and, at s==1, the initial P) is now complete.
    wait_async0();

    // Prefetch next matrix straight into LDS while we compute this step.
    if (s + 1 < SEQ) issue_mat(Bn, s + 1);

    // --- P(32x32) x B(32x32) with 16x16x4 f32 WMMA ------------------------
    v8f c00 = {}, c01 = {}, c10 = {}, c11 = {};
#pragma unroll
    for (int k0 = 0; k0 < MAT; k0 += 4) {
      const int k = k0 + 2 * half;   // this lane's K for vgpr0; vgpr1 is k+1
      v2f a0, a1, b0, b1;
      // A fragments: A[m][k], m = tileM + l16, vgpr v -> k0 + v + 2*half
      a0[0] = P[l16 * PAD + k];
      a0[1] = P[l16 * PAD + k + 1];
      a1[0] = P[(16 + l16) * PAD + k];
      a1[1] = P[(16 + l16) * PAD + k + 1];
      // B fragments: B[k][n], n = tileN + l16, vgpr v -> k0 + v + 2*half
      b0[0] = Bc[k * PAD + l16];
      b0[1] = Bc[(k + 1) * PAD + l16];
      b1[0] = Bc[k * PAD + 16 + l16];
      b1[1] = Bc[(k + 1) * PAD + 16 + l16];

      c00 = wmma_f32(a0, b0, c00);
      c01 = wmma_f32(a0, b1, c01);
      c10 = wmma_f32(a1, b0, c10);
      c11 = wmma_f32(a1, b1, c11);
    }

    // --- write product back to P (C/D layout: vgpr r -> M = tileM+r+8*half)
#pragma unroll
    for (int r = 0; r < 8; ++r) {
      P[(r + 8 * half) * PAD + l16]             = c00[r];
      P[(r + 8 * half) * PAD + 16 + l16]        = c01[r];
      P[(16 + r + 8 * half) * PAD + l16]        = c10[r];
      P[(16 + r + 8 * half) * PAD + 16 + l16]   = c11[r];
    }
  }

  // --- emit final 28x28 product as 784 contiguous floats ------------------
  float* orow = out + (size_t)b * EMB;
  for (int e4 = lane; e4 < NV4; e4 += 32) {
    int r  = e4 / ROWV4;
    int c4 = e4 % ROWV4;
    f4 v = *(const f4*)&P[r * PAD + 4 * c4];
    *(f4*)&orow[4 * e4] = v;
  }
}

extern "C" void kernel_launch(void* const* d_in, const int* in_sizes, int n_in,
                              void* d_out, int out_size, void* d_ws, size_t ws_size,
                              hipStream_t stream) {
  (void)in_sizes; (void)n_in; (void)out_size; (void)d_ws; (void)ws_size;
  const int*   sent = (const int*)d_in[0];     // (256, 64) int32 word ids
  const float* lw   = (const float*)d_in[1];   // (30001, 784) float32
  float*       out  = (float*)d_out;           // (256, 784) float32

  dim3 grid(BATCH / WPB);   // 64 blocks
  dim3 block(WPB * 32);     // 4 waves of 32
  hipLaunchKernelGGL(w2m_chain_kernel, grid, block, 0, stream, sent, lw, out);
}